// CGCNNEncoder_23063974379748
// MI455X (gfx1250) — compile-verified
//
#include <hip/hip_runtime.h>
#include <hip/hip_bf16.h>
#include <math.h>

// ---------------------------------------------------------------------------
// CGCNN encoder for MI455X (gfx1250), fp32 end-to-end.
// Main GEMMs use V_WMMA_F32_16X16X4_F32 (fp32 WMMA, full reference precision).
// BatchNorm (training mode) forces two passes over Z = tot@W+b; we RECOMPUTE
// the GEMM instead of materializing the 200MB Z buffer (HBM is the constraint,
// 58 GFLOP fp32 is ~free on this chip; atom_fea + weights are L2-resident).
// ---------------------------------------------------------------------------

typedef __attribute__((ext_vector_type(2))) float v2f;
typedef __attribute__((ext_vector_type(8))) float v8f;

#define NUM_B   1024
#define NPG     32
#define KNB     12
#define NATOM   (NUM_B*NPG)        // 32768
#define NEDGE   (NATOM*KNB)        // 393216
#define AFL     64
#define KDIM    192                 // 2*afl + nfl
#define MDIM    128                 // 2*afl
#define NLAYER  3
#define BN_EPS  1e-5f
#define GSTEP   (8.0f/63.0f)
#define GCOEFF  (-0.5f/(GSTEP*GSTEP))
#define TOT_STRIDE 196              // %64 == 4 -> conflict-free A-frag LDS reads

__device__ __forceinline__ float softplus_f(float x) {
    return x > 20.0f ? x : log1pf(expf(x));
}
__device__ __forceinline__ float sigmoid_f(float x) {
    return 1.0f / (1.0f + expf(-x));
}

// ---------------------------------------------------------------------------
// Edge construction: per graph, 32x32 minimum-image distances, 12-NN per atom.
// Tie-break: strictly-less keeps the first (lowest) index, matching lax.top_k.
// ---------------------------------------------------------------------------
__global__ __launch_bounds__(32) void k_edges(
    const float* __restrict__ lattice, const float* __restrict__ fracs,
    int* __restrict__ edge_src, float* __restrict__ edge_dist) {
    __shared__ float f[NPG][3];
    __shared__ float lat[9];
    __shared__ float dmat[NPG][NPG + 1];
    int b = blockIdx.x, i = threadIdx.x;
    f[i][0] = fracs[(b * NPG + i) * 3 + 0];
    f[i][1] = fracs[(b * NPG + i) * 3 + 1];
    f[i][2] = fracs[(b * NPG + i) * 3 + 2];
    if (i < 9) lat[i] = lattice[b * 9 + i];
    __syncthreads();
    for (int j = 0; j < NPG; ++j) {
        float dx = f[i][0] - f[j][0]; dx -= rintf(dx);   // jnp.round = rint
        float dy = f[i][1] - f[j][1]; dy -= rintf(dy);
        float dz = f[i][2] - f[j][2]; dz -= rintf(dz);
        float cx = dx * lat[0] + dy * lat[3] + dz * lat[6];
        float cy = dx * lat[1] + dy * lat[4] + dz * lat[7];
        float cz = dx * lat[2] + dy * lat[5] + dz * lat[8];
        float sq = cx * cx + cy * cy + cz * cz;
        dmat[i][j] = (i == j) ? INFINITY : sqrtf(sq);
    }
    // selection of 12 smallest (no cross-thread sharing of dmat rows)
    for (int kk = 0; kk < KNB; ++kk) {
        float best = INFINITY; int bj = 0;
        for (int j = 0; j < NPG; ++j) {
            float d = dmat[i][j];
            if (d < best) { best = d; bj = j; }
        }
        dmat[i][bj] = INFINITY;
        int e = (b * NPG + i) * KNB + kk;
        edge_src[e]  = b * NPG + bj;
        edge_dist[e] = best;
    }
}

// atom_fea = embedding[species]
__global__ __launch_bounds__(256) void k_init_atom(
    const int* __restrict__ species, const float* __restrict__ embedding,
    float* __restrict__ af) {
    int idx = blockIdx.x * 256 + threadIdx.x;
    if (idx < NATOM * AFL) {
        int a = idx >> 6, c = idx & 63;
        af[idx] = embedding[species[a] * AFL + c];
    }
}

__global__ void k_zero(float* __restrict__ p) { p[threadIdx.x] = 0.0f; }

// ---------------------------------------------------------------------------
// Pass 1: Z = tot @ W + b (WMMA fp32), accumulate per-column sum / sumsq.
// Block = 4 waves; each wave owns 48 edges (3 row-tiles of 16).
// ---------------------------------------------------------------------------
__global__ __launch_bounds__(128) void k_gemm_stats(
    const float* __restrict__ af, const int* __restrict__ edge_src,
    const float* __restrict__ edge_dist, const float* __restrict__ W,
    const float* __restrict__ bias, float* __restrict__ accum1) {
    __shared__ float tot[4][16 * TOT_STRIDE];
    __shared__ int   srcIdx[4][16];
    __shared__ float sAcc[MDIM], qAcc[MDIM];

    int tid  = threadIdx.x;
    int wid  = tid >> 5, lane = tid & 31;
    int lm   = lane & 15, hk = lane >> 4;
    int wE0  = (blockIdx.x * 4 + wid) * 48;
    float* tw = &tot[wid][0];

    if (tid < MDIM) { sAcc[tid] = 0.0f; qAcc[tid] = 0.0f; }
    __syncthreads();

    const v8f vzero = {0.f, 0.f, 0.f, 0.f, 0.f, 0.f, 0.f, 0.f};
    for (int t = 0; t < 3; ++t) {
        int eBase = wE0 + t * 16;
        if (lane < 16) srcIdx[wid][lane] = edge_src[eBase + lane];
        __syncthreads();
        for (int idx = lane; idx < 16 * KDIM; idx += 32) {
            int r = idx / KDIM, c = idx - r * KDIM;
            int e = eBase + r;
            float v;
            if (c < 64)       v = af[srcIdx[wid][r] * AFL + c];
            else if (c < 128) v = af[(e / KNB) * AFL + (c - 64)];
            else {
                float u = edge_dist[e] - GSTEP * (float)(c - 128);
                v = expf(GCOEFF * u * u);
            }
            tw[r * TOT_STRIDE + c] = v;
        }
        __syncthreads();

        v8f acc[8];
        #pragma unroll
        for (int nt = 0; nt < 8; ++nt) acc[nt] = vzero;
        for (int kk = 0; kk < 48; ++kk) {
            int k0 = kk * 4 + 2 * hk;
            v2f a;
            a.x = tw[lm * TOT_STRIDE + k0];
            a.y = tw[lm * TOT_STRIDE + k0 + 1];
            #pragma unroll
            for (int nt = 0; nt < 8; ++nt) {
                v2f bf;
                bf.x = W[k0 * MDIM + nt * 16 + lm];
                bf.y = W[(k0 + 1) * MDIM + nt * 16 + lm];
                acc[nt] = __builtin_amdgcn_wmma_f32_16x16x4_f32(
                    false, a, false, bf, (short)0, acc[nt], false, false);
            }
        }
        // column sum / sumsq over this 16-row tile (D layout: M = v + 8*hk, N = lm)
        #pragma unroll
        for (int nt = 0; nt < 8; ++nt) {
            int n = nt * 16 + lm;
            float bb = bias[n];
            float s = 0.0f, q = 0.0f;
            #pragma unroll
            for (int v = 0; v < 8; ++v) {
                float z = acc[nt][v] + bb;
                s += z; q += z * z;
            }
            s += __shfl_xor(s, 16, 32);
            q += __shfl_xor(q, 16, 32);
            if (lane < 16) { atomicAdd(&sAcc[n], s); atomicAdd(&qAcc[n], q); }
        }
        __syncthreads();
    }
    if (tid < MDIM) {
        atomicAdd(&accum1[tid], sAcc[tid]);
        atomicAdd(&accum1[MDIM + tid], qAcc[tid]);
    }
}

// accum1 -> per-column affine (scale, shift) folding BN1
__global__ void k_fin1(const float* __restrict__ accum1,
                       const float* __restrict__ g, const float* __restrict__ b,
                       float* __restrict__ bn1p) {
    int c = threadIdx.x;                       // 128
    float mean = accum1[c] * (1.0f / NEDGE);
    float var  = accum1[MDIM + c] * (1.0f / NEDGE) - mean * mean;
    float sc   = g[c] * rsqrtf(var + BN_EPS);
    bn1p[c] = sc;
    bn1p[MDIM + c] = b[c] - mean * sc;
}

// ---------------------------------------------------------------------------
// Pass 2: recompute Z, apply BN1, msg = sigmoid(filt)*softplus(core),
// exact segment-sum (each wave owns 4 dst atoms = 48 contiguous edges).
// ---------------------------------------------------------------------------
__global__ __launch_bounds__(128) void k_gemm_msg(
    const float* __restrict__ af, const int* __restrict__ edge_src,
    const float* __restrict__ edge_dist, const float* __restrict__ W,
    const float* __restrict__ bias, const float* __restrict__ bn1p,
    float* __restrict__ upd) {
    __shared__ float tot[4][16 * TOT_STRIDE];
    __shared__ float msg[4][48 * AFL];
    __shared__ int   srcIdx[4][16];

    int tid  = threadIdx.x;
    int wid  = tid >> 5, lane = tid & 31;
    int lm   = lane & 15, hk = lane >> 4;
    int wE0  = (blockIdx.x * 4 + wid) * 48;
    float* tw = &tot[wid][0];
    float* mw = &msg[wid][0];

    const v8f vzero = {0.f, 0.f, 0.f, 0.f, 0.f, 0.f, 0.f, 0.f};
    for (int t = 0; t < 3; ++t) {
        int eBase = wE0 + t * 16;
        if (lane < 16) srcIdx[wid][lane] = edge_src[eBase + lane];
        __syncthreads();
        for (int idx = lane; idx < 16 * KDIM; idx += 32) {
            int r = idx / KDIM, c = idx - r * KDIM;
            int e = eBase + r;
            float v;
            if (c < 64)       v = af[srcIdx[wid][r] * AFL + c];
            else if (c < 128) v = af[(e / KNB) * AFL + (c - 64)];
            else {
                float u = edge_dist[e] - GSTEP * (float)(c - 128);
                v = expf(GCOEFF * u * u);
            }
            tw[r * TOT_STRIDE + c] = v;
        }
        __syncthreads();

        v8f acc[8];
        #pragma unroll
        for (int nt = 0; nt < 8; ++nt) acc[nt] = vzero;
        for (int kk = 0; kk < 48; ++kk) {
            int k0 = kk * 4 + 2 * hk;
            v2f a;
            a.x = tw[lm * TOT_STRIDE + k0];
            a.y = tw[lm * TOT_STRIDE + k0 + 1];
            #pragma unroll
            for (int nt = 0; nt < 8; ++nt) {
                v2f bf;
                bf.x = W[k0 * MDIM + nt * 16 + lm];
                bf.y = W[(k0 + 1) * MDIM + nt * 16 + lm];
                acc[nt] = __builtin_amdgcn_wmma_f32_16x16x4_f32(
                    false, a, false, bf, (short)0, acc[nt], false, false);
            }
        }
        // BN1 + gated message into LDS
        #pragma unroll
        for (int nt = 0; nt < 4; ++nt) {
            int nf = nt * 16 + lm, nc = nf + 64;
            float bF = bias[nf], bC = bias[nc];
            float sF = bn1p[nf], hF = bn1p[MDIM + nf];
            float sC = bn1p[nc], hC = bn1p[MDIM + nc];
            #pragma unroll
            for (int v = 0; v < 8; ++v) {
                float zf = (acc[nt][v] + bF) * sF + hF;
                float zc = (acc[nt + 4][v] + bC) * sC + hC;
                int row = t * 16 + v + 8 * hk;
                mw[row * AFL + nf] = sigmoid_f(zf) * softplus_f(zc);
            }
        }
        __syncthreads();
    }
    // segment sum: 4 atoms x 64 features, groups of 12 contiguous edges
    int atom0 = wE0 / KNB;
    for (int q = lane; q < 4 * AFL; q += 32) {
        int a = q >> 6, c = q & 63;
        float s = 0.0f;
        #pragma unroll
        for (int r = 0; r < KNB; ++r) s += mw[(a * KNB + r) * AFL + c];
        upd[(atom0 + a) * AFL + c] = s;
    }
}

// upd column stats for BN2
__global__ __launch_bounds__(64) void k_stats_upd(
    const float* __restrict__ upd, float* __restrict__ accum2) {
    int c = threadIdx.x;
    int r0 = blockIdx.x * (NATOM / 128);
    float s = 0.0f, q = 0.0f;
    for (int r = 0; r < NATOM / 128; ++r) {
        float v = upd[(r0 + r) * AFL + c];
        s += v; q += v * v;
    }
    atomicAdd(&accum2[c], s);
    atomicAdd(&accum2[AFL + c], q);
}

__global__ void k_fin2(const float* __restrict__ accum2,
                       const float* __restrict__ g, const float* __restrict__ b,
                       float* __restrict__ bn2p) {
    int c = threadIdx.x;                       // 64
    float mean = accum2[c] * (1.0f / NATOM);
    float var  = accum2[AFL + c] * (1.0f / NATOM) - mean * mean;
    float sc   = g[c] * rsqrtf(var + BN_EPS);
    bn2p[c] = sc;
    bn2p[AFL + c] = b[c] - mean * sc;
}

// atom_fea = softplus(atom_fea + BN2(upd))
__global__ __launch_bounds__(256) void k_update(
    const float* __restrict__ upd, const float* __restrict__ bn2p,
    float* __restrict__ af) {
    int idx = blockIdx.x * 256 + threadIdx.x;
    if (idx < NATOM * AFL) {
        int c = idx & 63;
        float u = upd[idx] * bn2p[c] + bn2p[AFL + c];
        af[idx] = softplus_f(af[idx] + u);
    }
}

// crys = [mean-pooled atom_fea | lattice]  ([B,73])
__global__ __launch_bounds__(128) void k_pool(
    const float* __restrict__ af, const float* __restrict__ lattice,
    float* __restrict__ crys) {
    int b = blockIdx.x, t = threadIdx.x;
    if (t < AFL) {
        float s = 0.0f;
        #pragma unroll 4
        for (int a = 0; a < NPG; ++a) s += af[(b * NPG + a) * AFL + t];
        crys[b * 73 + t] = s * (1.0f / NPG);
    } else if (t < 73) {
        crys[b * 73 + t] = lattice[b * 9 + (t - AFL)];
    }
}

__global__ __launch_bounds__(128) void k_head1(
    const float* __restrict__ crys, const float* __restrict__ W1,
    const float* __restrict__ b1, float* __restrict__ h) {
    __shared__ float cl[73];
    int b = blockIdx.x, t = threadIdx.x;
    if (t < 73) cl[t] = crys[b * 73 + t];
    __syncthreads();
    float acc = b1[t];
    for (int c = 0; c < 73; ++c) acc += cl[c] * W1[c * 128 + t];
    h[b * 128 + t] = softplus_f(acc);
}

__global__ __launch_bounds__(256) void k_head2(
    const float* __restrict__ h, const float* __restrict__ W2,
    const float* __restrict__ b2, float* __restrict__ out) {
    __shared__ float hl[128];
    int b = blockIdx.x, t = threadIdx.x;
    if (t < 128) hl[t] = h[b * 128 + t];
    __syncthreads();
    float acc = b2[t];
    for (int c = 0; c < 128; ++c) acc += hl[c] * W2[c * 256 + t];
    if (t < 128) out[b * 128 + t] = acc;                         // mu
    else         out[NUM_B * 128 + b * 128 + (t - 128)] = acc;   // logvar
}

// ---------------------------------------------------------------------------
extern "C" void kernel_launch(void* const* d_in, const int* in_sizes, int n_in,
                              void* d_out, int out_size, void* d_ws, size_t ws_size,
                              hipStream_t stream) {
    (void)in_sizes; (void)n_in; (void)out_size; (void)ws_size;
    const float* lattice   = (const float*)d_in[0];
    const float* fracs     = (const float*)d_in[1];
    const int*   species   = (const int*)d_in[2];
    // d_in[3] batch_indices, d_in[4] n_per_graph, d_in[5] k_neighbors: implicit
    const float* embedding = (const float*)d_in[6];
    const float* fc_W      = (const float*)d_in[7];
    const float* fc_b      = (const float*)d_in[8];
    const float* bn1_g     = (const float*)d_in[9];
    const float* bn1_b     = (const float*)d_in[10];
    const float* bn2_g     = (const float*)d_in[11];
    const float* bn2_b     = (const float*)d_in[12];
    const float* head_W1   = (const float*)d_in[13];
    const float* head_b1   = (const float*)d_in[14];
    const float* head_W2   = (const float*)d_in[15];
    const float* head_b2   = (const float*)d_in[16];
    float* out = (float*)d_out;

    // workspace layout (~21 MB of floats)
    float* ws        = (float*)d_ws;
    float* edge_dist = ws;                              // NEDGE
    int*   edge_src  = (int*)(ws + NEDGE);              // NEDGE
    float* af        = ws + 2 * NEDGE;                  // NATOM*64
    float* upd       = af + NATOM * AFL;                // NATOM*64
    float* accum1    = upd + NATOM * AFL;               // 256
    float* accum2    = accum1 + 256;                    // 128
    float* bn1p      = accum2 + 128;                    // 256
    float* bn2p      = bn1p + 256;                      // 128
    float* crys      = bn2p + 128;                      // B*73
    float* hbuf      = crys + NUM_B * 73;               // B*128

    k_edges<<<NUM_B, 32, 0, stream>>>(lattice, fracs, edge_src, edge_dist);
    k_init_atom<<<(NATOM * AFL) / 256, 256, 0, stream>>>(species, embedding, af);

    for (int l = 0; l < NLAYER; ++l) {
        const float* Wl  = fc_W + (size_t)l * KDIM * MDIM;
        const float* bl  = fc_b + l * MDIM;
        k_zero<<<1, 384, 0, stream>>>(accum1);          // zeros accum1+accum2
        k_gemm_stats<<<NEDGE / 192, 128, 0, stream>>>(af, edge_src, edge_dist,
                                                      Wl, bl, accum1);
        k_fin1<<<1, MDIM, 0, stream>>>(accum1, bn1_g + l * MDIM, bn1_b + l * MDIM, bn1p);
        k_gemm_msg<<<NEDGE / 192, 128, 0, stream>>>(af, edge_src, edge_dist,
                                                    Wl, bl, bn1p, upd);
        k_stats_upd<<<128, 64, 0, stream>>>(upd, accum2);
        k_fin2<<<1, AFL, 0, stream>>>(accum2, bn2_g + l * AFL, bn2_b + l * AFL, bn2p);
        k_update<<<(NATOM * AFL) / 256, 256, 0, stream>>>(upd, bn2p, af);
    }

    k_pool<<<NUM_B, 128, 0, stream>>>(af, lattice, crys);
    k_head1<<<NUM_B, 128, 0, stream>>>(crys, head_W1, head_b1, hbuf);
    k_head2<<<NUM_B, 256, 0, stream>>>(hbuf, head_W2, head_b2, out);
}